// FNOFactorizedBlock1D_42984032698426
// MI455X (gfx1250) — compile-verified
//
#include <hip/hip_runtime.h>
#include <hip/hip_bf16.h>

// ---------------------------------------------------------------------------
// FNO factorized block, CDNA5 (gfx1250), fp32 via V_WMMA_F32_16X16X4_F32.
//
// B=16, S=8192, WIDTH=64, MODES=16, NL=4, HID=128, IN=12, OUT=5.
// Spectral conv == dense DFT to 16 modes:  F = H @ Phi   (K = 8192)
//                  mode mix (tiny complex 64x64 per mode)
//                  H += G @ Psi            (N = 8192, K = 32, irfft scaling
//                                           folded into Psi, Im X0 dropped)
// ~11 GFLOP total, working set fits in L2 -> fp32 matrix cores are the right
// precision (long K=8192 reductions; bf16 would lose accuracy for zero gain).
// ---------------------------------------------------------------------------

typedef float v2f __attribute__((ext_vector_type(2)));
typedef float v8f __attribute__((ext_vector_type(8)));

__device__ __forceinline__ v8f wmma4(v2f a, v2f b, v8f c) {
    // D = A(16x4,f32) * B(4x16,f32) + C(16x16,f32)
    return __builtin_amdgcn_wmma_f32_16x16x4_f32(
        /*neg_a=*/false, a, /*neg_b=*/false, b,
        /*c_mod=*/(short)0, c, /*reuse_a=*/false, /*reuse_b=*/false);
}

#define S_LEN   8192
#define NCH     64
#define NROWS   1024          /* B*WIDTH */
#define NMODE2  32            /* 16 real + 16 imag columns */

// ---------------------------------------------------------------------------
// Basis generation: fwd[s*32+k] (Phi, row-major by s), inv[k*8192+s] (Psi).
// ---------------------------------------------------------------------------
__global__ __launch_bounds__(256) void k_basis(float* __restrict__ fwd,
                                               float* __restrict__ inv) {
    int gid = blockIdx.x * 256 + threadIdx.x;       // 0 .. 8192*32-1
    int s = gid >> 5, k = gid & 31, km = k & 15;
    long prod = (long)km * (long)s;
    double theta = (double)(prod & (S_LEN - 1)) *
                   (6.283185307179586476925286766559005768 / (double)S_LEN);
    float c = (float)cos(theta);
    float sn = (float)sin(theta);
    const float invS = 1.0f / (float)S_LEN;
    // rfft: Re = sum h*cos ; Im = -sum h*sin
    fwd[s * 32 + k] = (k < 16) ? c : -sn;
    // irfft with only modes 0..15: out = (1/S)[ReX0 + 2*sum(ReXk cos - ImXk sin)]
    float v;
    if (k < 16) v = ((km == 0) ? invS : 2.0f * invS) * c;
    else        v = ((km == 0) ? 0.0f : (-2.0f * invS)) * sn;  // ImX0 dropped
    inv[(long)k * S_LEN + s] = v;
}

// ---------------------------------------------------------------------------
// Input projection: H[b*64+c][s] = b_in[c] + sum_j x[b,s,j] * w_in[j,c]
// ---------------------------------------------------------------------------
__global__ __launch_bounds__(256) void k_inproj(const float* __restrict__ x,
                                                const float* __restrict__ w_in,
                                                const float* __restrict__ b_in,
                                                float* __restrict__ H) {
    __shared__ float W[12 * 64];
    __shared__ float Bv[64];
    int t = threadIdx.x;
    for (int i = t; i < 12 * 64; i += 256) W[i] = w_in[i];
    if (t < 64) Bv[t] = b_in[t];
    __syncthreads();
    int gid = blockIdx.x * 256 + t;                 // (b,s) flat, 131072 total
    int b = gid >> 13, s = gid & (S_LEN - 1);
    float xv[12];
#pragma unroll
    for (int j = 0; j < 12; ++j) xv[j] = x[(long)gid * 12 + j];
    for (int c = 0; c < 64; ++c) {
        float acc = Bv[c];
#pragma unroll
        for (int j = 0; j < 12; ++j) acc += xv[j] * W[j * 64 + c];
        H[((long)b * 64 + c) * S_LEN + s] = acc;    // coalesced over s
    }
}

// ---------------------------------------------------------------------------
// Forward DFT: F(1024x32) = H(1024x8192) @ Phi(8192x32).
// One block per 16-row M tile; 8 waves = 2 N-tiles x 4 K-chunks, LDS reduce.
// ---------------------------------------------------------------------------
__global__ __launch_bounds__(256) void k_fwd_dft(const float* __restrict__ H,
                                                 const float* __restrict__ fwdB,
                                                 float* __restrict__ F) {
    __shared__ float red[8][32][8];                 // 8 KB
    const int t = threadIdx.x;
    const int w = t >> 5, lane = t & 31, lr = lane & 15, lh = lane >> 4;
    const int mt = blockIdx.x;                      // 0..63
    const int nt = w & 1;                           // N tile (cos | -sin cols)
    const int kc = w >> 1;                          // K chunk 0..3
    const float* Arow = H + ((long)mt * 16 + lr) * S_LEN;
    const float* Bcol = fwdB + nt * 16 + lr;
    v8f acc = {0.f, 0.f, 0.f, 0.f, 0.f, 0.f, 0.f, 0.f};
    int k0 = kc * 2048, kend = k0 + 2048;
    for (; k0 < kend; k0 += 4) {
        v2f a = *(const v2f*)(Arow + k0 + 2 * lh);  // A: 16x4 fragment
        v2f b;
        b.x = Bcol[(k0 + 2 * lh) * 32];             // B: 4x16 fragment
        b.y = Bcol[(k0 + 2 * lh + 1) * 32];
        acc = wmma4(a, b, acc);
    }
#pragma unroll
    for (int j = 0; j < 8; ++j) red[w][lane][j] = acc[j];
    __syncthreads();
    if (w < 2) {                                    // reduce 4 K-chunks
#pragma unroll
        for (int j = 0; j < 8; ++j) {
            float s = red[w][lane][j] + red[w + 2][lane][j] +
                      red[w + 4][lane][j] + red[w + 6][lane][j];
            F[((long)mt * 16 + 8 * lh + j) * 32 + w * 16 + lr] = s;
        }
    }
}

// ---------------------------------------------------------------------------
// Mode mix: per mode k, complex  G[b,o] = sum_i F[b,i] * (Wr + i*Wi)[i,o,k].
// Tiny (8 MFLOP/layer) -> plain VALU, one block per mode.
// ---------------------------------------------------------------------------
__global__ __launch_bounds__(256) void k_modemix(const float* __restrict__ F,
                                                 const float* __restrict__ Wr,
                                                 const float* __restrict__ Wi,
                                                 float* __restrict__ G) {
    __shared__ float Fr[NROWS], Fi[NROWS];
    const int k = blockIdx.x;                       // mode 0..15
    const int t = threadIdx.x;
    for (int idx = t; idx < NROWS; idx += 256) {
        Fr[idx] = F[idx * 32 + k];
        Fi[idx] = F[idx * 32 + 16 + k];
    }
    __syncthreads();
#pragma unroll
    for (int u = 0; u < 4; ++u) {
        int oid = t + u * 256;                      // (b,o) flat
        int b = oid >> 6, o = oid & 63;
        float gr = 0.f, gi = 0.f;
        for (int i = 0; i < 64; ++i) {
            float wr = Wr[((long)i * 64 + o) * 16 + k];
            float wi = Wi[((long)i * 64 + o) * 16 + k];
            float fr = Fr[b * 64 + i], fi = Fi[b * 64 + i];
            gr += fr * wr - fi * wi;
            gi += fr * wi + fi * wr;
        }
        G[((long)b * 64 + o) * 32 + k]      = gr;
        G[((long)b * 64 + o) * 32 + 16 + k] = gi;
    }
}

// ---------------------------------------------------------------------------
// Inverse + residual (in place): H(1024x8192) += G(1024x32) @ Psi(32x8192).
// Each wave owns one 16x16 tile; its H tile is read only as its own C init,
// so in-place update is race-free. 4096 blocks x 8 waves = 512 N-tiles/M-tile.
// ---------------------------------------------------------------------------
__global__ __launch_bounds__(256) void k_inv(float* __restrict__ H,
                                             const float* __restrict__ G,
                                             const float* __restrict__ invB) {
    const int t = threadIdx.x, w = t >> 5, lane = t & 31;
    const int lr = lane & 15, lh = lane >> 4;
    const int mt = blockIdx.x >> 6;                 // 0..63
    const int nt = (blockIdx.x & 63) * 8 + w;       // 0..511
    const int n0 = nt * 16;
    float* Hp = H + ((long)mt * 16) * S_LEN + n0;
    v8f acc;
#pragma unroll
    for (int j = 0; j < 8; ++j)                     // C = H tile (residual)
        acc[j] = Hp[(8 * lh + j) * S_LEN + lr];
    const float* Ga = G + ((long)mt * 16 + lr) * 32;
#pragma unroll
    for (int k0 = 0; k0 < 32; k0 += 4) {
        v2f a = *(const v2f*)(Ga + k0 + 2 * lh);
        v2f b;
        b.x = invB[(long)(k0 + 2 * lh) * S_LEN + n0 + lr];
        b.y = invB[(long)(k0 + 2 * lh + 1) * S_LEN + n0 + lr];
        acc = wmma4(a, b, acc);
    }
#pragma unroll
    for (int j = 0; j < 8; ++j)
        Hp[(8 * lh + j) * S_LEN + lr] = acc[j];
}

// ---------------------------------------------------------------------------
// Transpose H (B,64,S) -> h_out (B,S,64) (this IS output #3).
// ---------------------------------------------------------------------------
__global__ __launch_bounds__(256) void k_transpose(const float* __restrict__ H,
                                                   float* __restrict__ hout) {
    __shared__ float tile[32][33];
    int b  = blockIdx.x >> 9;
    int ct = (blockIdx.x >> 8) & 1;                 // channel tile (64/32)
    int st = blockIdx.x & 255;                      // s tile (8192/32)
    int tx = threadIdx.x & 31, ty = threadIdx.x >> 5;   // 32x8
    const float* src = H + ((long)b * 64 + ct * 32) * S_LEN + st * 32;
#pragma unroll
    for (int i = 0; i < 32; i += 8) tile[ty + i][tx] = src[(ty + i) * S_LEN + tx];
    __syncthreads();
    float* dst = hout + ((long)b * S_LEN + st * 32) * 64 + ct * 32;
#pragma unroll
    for (int i = 0; i < 32; i += 8) dst[(ty + i) * 64 + tx] = tile[tx][ty + i];
}

// ---------------------------------------------------------------------------
// Fused two-stage head:  out = (A @ W1 + B1) @ W2 + B2
// A = h_out (131072 x 64), W1 (64x128), W2 (128xN2), N2 in {5, 64}.
// Block = 64 rows; stage1 -> LDS T(64x128, padded), stage2 from LDS.
// Bias rides in the WMMA C fragment. N2=5 handled via zero-padded LDS W2.
// ---------------------------------------------------------------------------
template <int N2>
__global__ __launch_bounds__(256) void k_head(const float* __restrict__ A,
                                              const float* __restrict__ W1,
                                              const float* __restrict__ B1,
                                              const float* __restrict__ W2,
                                              const float* __restrict__ B2,
                                              float* __restrict__ out) {
    constexpr int NT2 = (N2 + 15) / 16;
    constexpr bool PAD = (N2 % 16) != 0;
    __shared__ float T[64][132];                    // 33.8 KB
    __shared__ float W2p[PAD ? 128 * 16 : 1];       // 8 KB when padding
    const int t = threadIdx.x, w = t >> 5, lane = t & 31;
    const int lr = lane & 15, lh = lane >> 4;
    const long r0 = (long)blockIdx.x * 64;

    if constexpr (PAD) {
        for (int i = t; i < 128 * 16; i += 256) {
            int kk = i >> 4, nn = i & 15;
            W2p[i] = (nn < N2) ? W2[kk * N2 + nn] : 0.0f;
        }
    }

    // ---- stage 1: T = A[r0:r0+64] @ W1 + B1 ; wave w owns N-tile w ----
    {
        const int n0 = w * 16;
        const float bias = B1[n0 + lr];
        for (int mt = 0; mt < 4; ++mt) {
            v8f acc;
#pragma unroll
            for (int j = 0; j < 8; ++j) acc[j] = bias;
            const float* Ar = A + (r0 + mt * 16 + lr) * 64;
#pragma unroll
            for (int k0 = 0; k0 < 64; k0 += 4) {
                v2f a = *(const v2f*)(Ar + k0 + 2 * lh);
                v2f b;
                b.x = W1[(k0 + 2 * lh) * 128 + n0 + lr];
                b.y = W1[(k0 + 2 * lh + 1) * 128 + n0 + lr];
                acc = wmma4(a, b, acc);
            }
#pragma unroll
            for (int j = 0; j < 8; ++j) T[mt * 16 + 8 * lh + j][n0 + lr] = acc[j];
        }
    }
    __syncthreads();

    // ---- stage 2: out = T @ W2 + B2 (wave-uniform tile loop) ----
    for (int idx = w; idx < 4 * NT2; idx += 8) {
        const int mt = idx & 3, nt = idx >> 2;
        const int n0 = nt * 16;
        v8f acc;
        const float bias = (n0 + lr < N2) ? B2[n0 + lr] : 0.0f;
#pragma unroll
        for (int j = 0; j < 8; ++j) acc[j] = bias;
#pragma unroll 8
        for (int k0 = 0; k0 < 128; k0 += 4) {
            v2f a;
            a.x = T[mt * 16 + lr][k0 + 2 * lh];
            a.y = T[mt * 16 + lr][k0 + 2 * lh + 1];
            v2f b;
            if constexpr (PAD) {
                b.x = W2p[(k0 + 2 * lh) * 16 + lr];
                b.y = W2p[(k0 + 2 * lh + 1) * 16 + lr];
            } else {
                b.x = W2[(k0 + 2 * lh) * N2 + n0 + lr];
                b.y = W2[(k0 + 2 * lh + 1) * N2 + n0 + lr];
            }
            acc = wmma4(a, b, acc);
        }
#pragma unroll
        for (int j = 0; j < 8; ++j) {
            const int n = n0 + lr;
            if (n < N2)
                out[(r0 + mt * 16 + 8 * lh + j) * N2 + n] = acc[j];
        }
    }
}

// ---------------------------------------------------------------------------
extern "C" void kernel_launch(void* const* d_in, const int* in_sizes, int n_in,
                              void* d_out, int out_size, void* d_ws, size_t ws_size,
                              hipStream_t stream) {
    const float* x       = (const float*)d_in[0];
    const float* w_in    = (const float*)d_in[1];
    const float* b_in    = (const float*)d_in[2];
    const float* spec_wr = (const float*)d_in[3];
    const float* spec_wi = (const float*)d_in[4];
    const float* w_out1  = (const float*)d_in[5];
    const float* b_out1  = (const float*)d_in[6];
    const float* w_out2  = (const float*)d_in[7];
    const float* b_out2  = (const float*)d_in[8];
    const float* w_res1  = (const float*)d_in[9];
    const float* b_res1  = (const float*)d_in[10];
    const float* w_res2  = (const float*)d_in[11];
    const float* b_res2  = (const float*)d_in[12];

    float* out_f   = (float*)d_out;                  // (16,8192,1,5)
    float* out_res = out_f + 16L * 8192 * 5;         // (16,8192,64)
    float* out_h   = out_res + 16L * 8192 * 64;      // (16,8192,64)

    float* ws   = (float*)d_ws;
    float* H    = ws;                                // 1024 x 8192
    float* fwdB = H + (long)NROWS * S_LEN;           // 8192 x 32
    float* invB = fwdB + (long)S_LEN * NMODE2;       // 32 x 8192
    float* F    = invB + (long)NMODE2 * S_LEN;       // 1024 x 32
    float* G    = F + (long)NROWS * NMODE2;          // 1024 x 32

    k_basis<<<1024, 256, 0, stream>>>(fwdB, invB);
    k_inproj<<<512, 256, 0, stream>>>(x, w_in, b_in, H);
    for (int l = 0; l < 4; ++l) {
        const long off = (long)l * 64 * 64 * 16;
        k_fwd_dft<<<64, 256, 0, stream>>>(H, fwdB, F);
        k_modemix<<<16, 256, 0, stream>>>(F, spec_wr + off, spec_wi + off, G);
        k_inv<<<4096, 256, 0, stream>>>(H, G, invB);
    }
    k_transpose<<<8192, 256, 0, stream>>>(H, out_h);
    k_head<5><<<2048, 256, 0, stream>>>(out_h, w_out1, b_out1, w_out2, b_out2, out_f);
    k_head<64><<<2048, 256, 0, stream>>>(out_h, w_res1, b_res1, w_res2, b_res2, out_res);
}